// Logic_41669772706486
// MI455X (gfx1250) — compile-verified
//
#include <hip/hip_runtime.h>
#include <hip/hip_bf16.h>

// ---------------------------------------------------------------------------
// Problem constants (match reference)
// ---------------------------------------------------------------------------
#define BATCH   2048
#define IN_DIM  16384
#define OUT_DIM 16384
#define ROWS_PER_THREAD 8

typedef __attribute__((ext_vector_type(2))) float v2f;
typedef __attribute__((ext_vector_type(8))) float v8f;

// COEFF (16 x 4) zero-padded to 16 x 16 so the WMMA B operand (4x16, cols
// 4..15 zero) can be loaded unconditionally -- no divergent select, no EXEC
// save/restore inside the WMMA loop.
__device__ __constant__ float c_coeff_pad[16][16] = {
    {0, 0, 0, 0},  {0, 0, 0, 1},  {0, 1, 0, -1}, {0, 1, 0, 0},
    {0, 0, 1, -1}, {0, 0, 1, 0},  {0, 1, 1, -2}, {0, 1, 1, -1},
    {1, -1, -1, 1},{1, -1, -1, 2},{1, 0, -1, 0}, {1, 0, -1, 1},
    {1, -1, 0, 0}, {1, -1, 0, 1}, {1, 0, 0, -1}, {1, 0, 0, 0}};
// (rows are 16 wide; unspecified entries zero-initialize)

// ---------------------------------------------------------------------------
// Phase 1: W[j][0..3] = softmax(weights[j][0..15]) @ COEFF   via WMMA f32
//
// One wave32 per 16 output rows. Operands are loaded directly in the CDNA5
// V_WMMA_F32_16X16X4_F32 register layout (ISA 7.12.2, 32-bit A 16x4):
//   chunk c covers K = 4c..4c+3
//   lane L in [0,16):  v0 = A[M=L][K=4c+0],    v1 = A[M=L][K=4c+1]
//   lane L in [16,32): v0 = A[M=L-16][K=4c+2], v1 = A[M=L-16][K=4c+3]
// B (4x16, COEFF chunk zero-padded to 16 cols) mirrors this with N striped
// across lanes. Accumulate K=16 with 4 chained WMMAs.
// ---------------------------------------------------------------------------
__global__ __launch_bounds__(32) void compute_w_kernel(
    const float* __restrict__ weights,  // (OUT_DIM, 16)
    float* __restrict__ W)              // (OUT_DIM, 4)
{
    const int block16 = blockIdx.x;   // group of 16 rows
    const int lane    = threadIdx.x;  // 0..31
    const int m       = lane & 15;    // row within group
    const int hi      = lane >> 4;    // 0: K%4 in {0,1}; 1: K%4 in {2,3}
    const int row     = block16 * 16 + m;

    // Load 8 weights per lane in WMMA-A chunk layout, and the matching
    // (unconditional) COEFF B-operand values.
    float w[8];
    float bv[8];
    const int n = lane & 15;  // B/D column
#pragma unroll
    for (int c = 0; c < 4; ++c) {
        const int k = 4 * c + 2 * hi;
        w[2 * c + 0]  = weights[row * 16 + k + 0];
        w[2 * c + 1]  = weights[row * 16 + k + 1];
        bv[2 * c + 0] = c_coeff_pad[k + 0][n];
        bv[2 * c + 1] = c_coeff_pad[k + 1][n];
    }

    // Row softmax: each row's 16 values live in lanes {L, L^16} (8 each).
    float mx = w[0];
#pragma unroll
    for (int t = 1; t < 8; ++t) mx = fmaxf(mx, w[t]);
    mx = fmaxf(mx, __shfl_xor(mx, 16, 32));
    float s = 0.0f;
#pragma unroll
    for (int t = 0; t < 8; ++t) { w[t] = __expf(w[t] - mx); s += w[t]; }
    s += __shfl_xor(s, 16, 32);
    const float inv = __builtin_amdgcn_rcpf(s);  // v_rcp_f32
#pragma unroll
    for (int t = 0; t < 8; ++t) w[t] *= inv;

    // D = A(16x16 nw, K-chunked) x B(COEFF zero-padded 4x16) + D
    v8f acc = {0.f, 0.f, 0.f, 0.f, 0.f, 0.f, 0.f, 0.f};
#pragma unroll
    for (int c = 0; c < 4; ++c) {
        v2f A;
        A[0] = w[2 * c + 0];
        A[1] = w[2 * c + 1];
        v2f B;
        B[0] = bv[2 * c + 0];
        B[1] = bv[2 * c + 1];
        // 8 args: (neg_a, A, neg_b, B, c_mod, C, reuse_a, reuse_b)
        acc = __builtin_amdgcn_wmma_f32_16x16x4_f32(
            false, A, false, B, (short)0, acc, false, false);
    }

    // D layout: VGPR r -> (lanes 0-15: M=r, N=lane) (lanes 16-31: M=8+r, N=lane-16)
    // Store W[j][c] for c = 0..3 only.
    float* Wout = W + (size_t)block16 * 16 * 4;
    if (n < 4) {
#pragma unroll
        for (int r = 0; r < 8; ++r) {
            const int j = hi * 8 + r;
            Wout[j * 4 + n] = acc[r];
        }
    }
}

// ---------------------------------------------------------------------------
// Phase 2: bandwidth kernel.
// out[i][j] = W0 + W1*a + W2*b + W3*a*b,  a = x[i][ia[j]], b = x[i][ib[j]]
// One thread per output column j; idx/W cached in registers and amortized
// over ROWS_PER_THREAD batch rows. Coalesced W/idx/out.
//
// L2 policy (MI455X: 192 MB global L2, x = 134 MB): the gathers into x only
// run at full speed if x stays L2-resident, so the 134 MB output write
// stream is issued NON-TEMPORAL (store TH=NT, ISA sec 7.3) to keep it from
// evicting x. W/idx stay regular-temporal (reused by every batch-row block).
// ---------------------------------------------------------------------------
__global__ __launch_bounds__(256) void gate_gather_kernel(
    const float* __restrict__ x,          // (BATCH, IN_DIM)
    const float* __restrict__ W,          // (OUT_DIM, 4)
    const long long* __restrict__ idx_a,  // (OUT_DIM,) int64
    const long long* __restrict__ idx_b,  // (OUT_DIM,) int64
    float* __restrict__ out)              // (BATCH, OUT_DIM)
{
    const int jblocks = OUT_DIM / 256;
    const int j  = (blockIdx.x % jblocks) * 256 + threadIdx.x;
    const int i0 = (blockIdx.x / jblocks) * ROWS_PER_THREAD;

    const int ia = (int)idx_a[j];
    const int ib = (int)idx_b[j];
    const float4 w = ((const float4*)W)[j];

    const float* __restrict__ xa = x + ia;
    const float* __restrict__ xb = x + ib;

#pragma unroll
    for (int r = 0; r < ROWS_PER_THREAD; ++r) {
        const size_t i = (size_t)(i0 + r);
        const float a = xa[i * IN_DIM];
        const float b = xb[i * IN_DIM];
        const float v = fmaf(w.y, a, fmaf(w.z, b, fmaf(w.w, a * b, w.x)));
        __builtin_nontemporal_store(v, out + i * OUT_DIM + j);
    }
}

// ---------------------------------------------------------------------------
// Launch
// ---------------------------------------------------------------------------
extern "C" void kernel_launch(void* const* d_in, const int* in_sizes, int n_in,
                              void* d_out, int out_size, void* d_ws, size_t ws_size,
                              hipStream_t stream)
{
    const float*     x       = (const float*)d_in[0];     // (2048, 16384) f32
    const float*     weights = (const float*)d_in[1];     // (16384, 16) f32
    const long long* idx_a   = (const long long*)d_in[2]; // (16384,) i64
    const long long* idx_b   = (const long long*)d_in[3]; // (16384,) i64
    float*           out     = (float*)d_out;             // (2048, 16384) f32
    float*           W       = (float*)d_ws;              // 16384*4 f32 = 256 KB scratch

    (void)in_sizes; (void)n_in; (void)out_size; (void)ws_size;

    // Phase 1: per-row softmax + COEFF projection via v_wmma_f32_16x16x4_f32
    compute_w_kernel<<<OUT_DIM / 16, 32, 0, stream>>>(weights, W);

    // Phase 2: gather + fused gate evaluation (bandwidth-bound, NT output)
    const int grid = (OUT_DIM / 256) * (BATCH / ROWS_PER_THREAD);
    gate_gather_kernel<<<grid, 256, 0, stream>>>(x, W, idx_a, idx_b, out);
}